// post_process_82781199663626
// MI455X (gfx1250) — compile-verified
//
#include <hip/hip_runtime.h>
#include <stdint.h>

#define N_ANCH   150000
#define N_CLS    80
#define KMAX     300
#define KPAD     304          // padded to multiple of 16 for WMMA tiles
#define SUPW     10           // ceil(320/32) words of suppression bitmask
#define SCORE_THR 0.05f
#define IOU_THR   0.5f
#define IMGF      768.0f
#define NBINS     2048
#define CAP       1024        // candidate capacity (expected ~370)
#define TILE      4096        // score staging tile (floats) in LDS
#define BLOCK     1024

typedef __attribute__((ext_vector_type(2))) float        v2f;
typedef __attribute__((ext_vector_type(8))) float        v8f;
typedef __attribute__((ext_vector_type(4))) unsigned int v4u;
typedef __attribute__((ext_vector_type(8))) int          v8i;
typedef __attribute__((ext_vector_type(4))) int          v4i;

// ---------------------------------------------------------------------------
// Kernel 1: decode + clip boxes (RetinaNet BBoxTransform), memory bound.
// ---------------------------------------------------------------------------
__global__ void decode_kernel(const float* __restrict__ anchors,
                              const float* __restrict__ regr,
                              float* __restrict__ outBoxes) {
  int i = blockIdx.x * blockDim.x + threadIdx.x;
  if (i >= N_ANCH) return;
  // gentle read-ahead on the streaming inputs (global_prefetch_b8)
  __builtin_prefetch(anchors + (size_t)i * 4 + 4096, 0, 0);
  __builtin_prefetch(regr    + (size_t)i * 4 + 4096, 0, 0);

  float ax1 = anchors[i * 4 + 0], ay1 = anchors[i * 4 + 1];
  float ax2 = anchors[i * 4 + 2], ay2 = anchors[i * 4 + 3];
  float wa = ax2 - ax1, ha = ay2 - ay1;
  float cxa = ax1 + 0.5f * wa, cya = ay1 + 0.5f * ha;

  float dx = regr[i * 4 + 0] * 0.1f;
  float dy = regr[i * 4 + 1] * 0.1f;
  float dw = regr[i * 4 + 2] * 0.2f;
  float dh = regr[i * 4 + 3] * 0.2f;

  float cx = cxa + dx * wa;
  float cy = cya + dy * ha;
  float w  = __expf(dw) * wa;
  float h  = __expf(dh) * ha;

  float x1 = fminf(fmaxf(cx - 0.5f * w, 0.f), IMGF);
  float y1 = fminf(fmaxf(cy - 0.5f * h, 0.f), IMGF);
  float x2 = fminf(fmaxf(cx + 0.5f * w, 0.f), IMGF);
  float y2 = fminf(fmaxf(cy + 0.5f * h, 0.f), IMGF);

  outBoxes[i * 4 + 0] = x1;
  outBoxes[i * 4 + 1] = y1;
  outBoxes[i * 4 + 2] = x2;
  outBoxes[i * 4 + 3] = y2;
}

// ---------------------------------------------------------------------------
// TDM: gather one class's strided score column tile into LDS.
// 2D descriptor: tile_dim0=1 elem, tile_dim1=rows, tensor_dim0_stride=80.
// ---------------------------------------------------------------------------
__device__ __forceinline__ void tdm_load_rows(unsigned ldsByteOff,
                                              const float* gptr,
                                              unsigned rows) {
#if defined(__has_builtin) && __has_builtin(__builtin_amdgcn_tensor_load_to_lds)
  unsigned long long ga = (unsigned long long)(uintptr_t)gptr;
  v4u g0;
  g0.x = 1u;                                            // count=1, gather off
  g0.y = ldsByteOff;                                    // lds_addr
  g0.z = (unsigned)ga;                                  // global_addr[31:0]
  g0.w = (unsigned)((ga >> 32) & 0x01FFFFFFull) | (2u << 30); // addr[56:32]|type=2
  v8i g1;
  g1[0] = (int)(2u << 16);                              // data_size=4B, no mask
  g1[1] = (int)((unsigned)N_CLS << 16);                 // tensor_dim0 = 80
  g1[2] = (int)((rows & 0xFFFFu) << 16);                // dim0_hi=0 | tensor_dim1 lo
  g1[3] = (int)(((rows >> 16) & 0xFFFFu) | (1u << 16)); // tensor_dim1 hi | tile_dim0=1
  g1[4] = (int)(rows & 0xFFFFu);                        // tile_dim1=rows, tile_dim2=0
  g1[5] = (int)N_CLS;                                   // tensor_dim0_stride = 80
  g1[6] = 0;
  g1[7] = 0;
  v4i zz4 = {0, 0, 0, 0};
  v8i zz8 = {0, 0, 0, 0, 0, 0, 0, 0};
  __builtin_amdgcn_tensor_load_to_lds(g0, g1, zz4, zz4, zz8, 0);
#else
  // Fallback: lanes of this wave copy directly.
  float* lds = (float*)(void*)(uintptr_t)ldsByteOff;  // never taken on gfx1250
  for (unsigned r = threadIdx.x & 31u; r < rows; r += 32u)
    lds[r] = gptr[(size_t)r * N_CLS];
#endif
}

__device__ __forceinline__ void tdm_wait() {
#if defined(__has_builtin) && __has_builtin(__builtin_amdgcn_s_wait_tensorcnt)
  __builtin_amdgcn_s_wait_tensorcnt(0);
#endif
}

// ---------------------------------------------------------------------------
// Kernel 2: one block per class. Histogram top-K select + bitonic sort +
// WMMA-assisted IoU suppression matrix + greedy NMS + output.
// ---------------------------------------------------------------------------
__global__ __launch_bounds__(BLOCK) void select_nms_kernel(
    const float* __restrict__ cls, const float* __restrict__ boxes,
    float* __restrict__ outScores, int* __restrict__ outLabels,
    float* __restrict__ outBoxes) {
  __shared__ float              stage[TILE];            // 16 KB TDM staging
  __shared__ unsigned           hist[NBINS];            // 8 KB
  __shared__ unsigned long long sk[CAP];                // 8 KB candidate keys
  __shared__ float4             sbox[KPAD];             // top-K boxes
  __shared__ float              sarea[KPAD];
  __shared__ float              sscore[KPAD];
  __shared__ unsigned           ssup[KPAD * SUPW];      // 12 KB suppression bits
  __shared__ unsigned           skeep[SUPW];
  __shared__ unsigned           s_candCount;
  __shared__ int                s_cut;

  const int c    = blockIdx.x;
  const int tid  = threadIdx.x;
  const int lane = tid & 31;
  const int wave = tid >> 5;
  const float BIN_SCALE = (float)NBINS / (1.0f - SCORE_THR);

  // ---- Pass 1: per-class score histogram (TDM-staged strided gather) ----
  for (int u = tid; u < NBINS; u += BLOCK) hist[u] = 0u;
  if (tid == 0) { s_candCount = 0u; s_cut = 0; }
  __syncthreads();

  for (int base = 0; base < N_ANCH; base += TILE) {
    int rows = min(TILE, N_ANCH - base);
    int perWave = (rows + 31) >> 5;
    int r0 = wave * perWave;
    int wrows = min(perWave, rows - r0);
    if (wrows > 0)
      tdm_load_rows((unsigned)(size_t)&stage[r0],
                    cls + (size_t)(base + r0) * N_CLS + c, (unsigned)wrows);
    tdm_wait();
    __syncthreads();
    for (int t = tid; t < rows; t += BLOCK) {
      float s = stage[t];
      if (s > SCORE_THR) {
        int bin = (int)((s - SCORE_THR) * BIN_SCALE);
        bin = min(max(bin, 0), NBINS - 1);
        atomicAdd(&hist[bin], 1u);
      }
    }
    __syncthreads();
  }

  // ---- Find cut bin so that count(bins >= cut) >= KMAX ----
  if (tid == 0) {
    unsigned acc = 0; int cut = 0;
    for (int b = NBINS - 1; b >= 0; --b) {
      acc += hist[b];
      if (acc >= (unsigned)KMAX) { cut = b; break; }
    }
    s_cut = cut;
  }
  __syncthreads();
  const int cut = s_cut;

  // ---- Pass 2: gather candidates (score,index) as sortable 64-bit keys ----
  for (int base = 0; base < N_ANCH; base += TILE) {
    int rows = min(TILE, N_ANCH - base);
    int perWave = (rows + 31) >> 5;
    int r0 = wave * perWave;
    int wrows = min(perWave, rows - r0);
    if (wrows > 0)
      tdm_load_rows((unsigned)(size_t)&stage[r0],
                    cls + (size_t)(base + r0) * N_CLS + c, (unsigned)wrows);
    tdm_wait();
    __syncthreads();
    for (int t = tid; t < rows; t += BLOCK) {
      float s = stage[t];
      if (s > SCORE_THR) {
        int bin = (int)((s - SCORE_THR) * BIN_SCALE);
        bin = min(max(bin, 0), NBINS - 1);
        if (bin >= cut) {
          unsigned pos = atomicAdd(&s_candCount, 1u);
          if (pos < CAP) {
            unsigned idx = (unsigned)(base + t);
            // descending score, then ascending index; score>0 so bits monotone
            sk[pos] = ((unsigned long long)__float_as_uint(s) << 32) |
                      (unsigned long long)(0xFFFFFFFFu - idx);
          }
        }
      }
    }
    __syncthreads();
  }

  // ---- Pad + bitonic sort (descending), CAP == blockDim ----
  int cnt = (int)min(s_candCount, (unsigned)CAP);
  for (int t = tid; t < CAP; t += BLOCK)
    if (t >= cnt) sk[t] = 0ull;
  __syncthreads();
  for (int k = 2; k <= CAP; k <<= 1) {
    for (int j = k >> 1; j > 0; j >>= 1) {
      int p = tid ^ j;
      if (p > tid) {
        bool up = ((tid & k) == 0);      // "up" blocks sorted descending
        unsigned long long x = sk[tid], y = sk[p];
        bool sw = up ? (x < y) : (x > y);
        if (sw) { sk[tid] = y; sk[p] = x; }
      }
      __syncthreads();
    }
  }

  // ---- Materialize top-K boxes/areas/scores into LDS ----
  int n = min(cnt, KMAX);
  for (int t = tid; t < KPAD; t += BLOCK) {
    if (t < n) {
      unsigned long long key = sk[t];
      unsigned idx = 0xFFFFFFFFu - (unsigned)(key & 0xFFFFFFFFull);
      float4 bb = ((const float4*)boxes)[idx];
      sbox[t] = bb;
      sarea[t] = (bb.z - bb.x) * (bb.w - bb.y);
      sscore[t] = __uint_as_float((unsigned)(key >> 32));
    } else {
      sbox[t] = make_float4(0.f, 0.f, 0.f, 0.f);
      sarea[t] = 0.f;
      sscore[t] = -1.f;
    }
  }
  for (int u = tid; u < KPAD * SUPW; u += BLOCK) ssup[u] = 0u;
  if (tid < SUPW) {
    int lo = tid * 32;
    unsigned wv;
    if (n >= lo + 32)      wv = 0xFFFFFFFFu;
    else if (n <= lo)      wv = 0u;
    else                   wv = (1u << (n - lo)) - 1u;
    skeep[tid] = wv;                      // initial keep = score > THR
  }
  __syncthreads();

  // ---- Suppression matrix in 16x16 tiles; area_i + area_j on matrix unit:
  //      D = A*B with A=[area_i | 1], B=[1 ; area_j]  (V_WMMA_F32_16X16X4_F32)
  const int NT = KPAD / 16;               // 19
  for (int t5 = wave; t5 < NT * NT; t5 += 32) {
    int ti = t5 / NT, tj = t5 % NT;
    int i0 = ti * 16, j0 = tj * 16;
    if (j0 + 15 <= i0) continue;          // entirely below diagonal: no j>i
    // unconditional LDS reads, lane-select into WMMA operand layout
    int l15 = lane & 15;
    float ai = sarea[i0 + l15];
    float aj = sarea[j0 + l15];
    bool lo16 = lane < 16;
    v2f a, b;
    a.x = lo16 ? ai : 0.f;                // A: M=lane, K0=area_i
    a.y = lo16 ? 1.f : 0.f;               //            K1=1
    b.x = lo16 ? 1.f : 0.f;               // B: N=lane, K0 row = 1
    b.y = lo16 ? aj : 0.f;                //            K1 row = area_j
    v8f cz = {0.f, 0.f, 0.f, 0.f, 0.f, 0.f, 0.f, 0.f};
    v8f d = __builtin_amdgcn_wmma_f32_16x16x4_f32(
        false, a, false, b, (short)0, cz, false, false);
    int m_add = (lane >> 4) << 3;         // D layout: M = r + 8*(lane>=16)
    int nn = lane & 15;                   //           N = lane % 16
#pragma unroll
    for (int r = 0; r < 8; ++r) {
      int i = i0 + r + m_add;
      int j = j0 + nn;
      if (j > i) {
        float4 bi = sbox[i], bj = sbox[j];
        float lx = fmaxf(bi.x, bj.x), ly = fmaxf(bi.y, bj.y);
        float rx = fminf(bi.z, bj.z), ry = fminf(bi.w, bj.w);
        float iw = fmaxf(rx - lx, 0.f), ih = fmaxf(ry - ly, 0.f);
        float inter = iw * ih;
        // iou > thr  <=>  inter > thr*(sumArea - inter + eps); denom > 0
        float denom = d[r] - inter + 1e-8f;   // d[r] = area_i + area_j
        if (inter > IOU_THR * denom)
          atomicOr(&ssup[i * SUPW + (j >> 5)], 1u << (j & 31));
      }
    }
  }
  __syncthreads();

  // ---- Greedy NMS over bitmasks (exact mirror of reference fori_loop) ----
  if (tid == 0) {
    for (int i = 0; i < KMAX; ++i) {
      if ((skeep[i >> 5] >> (i & 31)) & 1u) {
#pragma unroll
        for (int w = 0; w < SUPW; ++w) skeep[w] &= ~ssup[i * SUPW + w];
      }
    }
  }
  __syncthreads();

  // ---- Emit outputs: scores / labels / boxes ----
  for (int t = tid; t < KMAX; t += BLOCK) {
    bool kept = (skeep[t >> 5] >> (t & 31)) & 1u;
    int o = c * KMAX + t;
    outScores[o] = kept ? sscore[t] : -1.f;
    outLabels[o] = c;
    float4 bb = kept ? sbox[t] : make_float4(0.f, 0.f, 0.f, 0.f);
    ((float4*)outBoxes)[o] = bb;
  }
}

// ---------------------------------------------------------------------------
extern "C" void kernel_launch(void* const* d_in, const int* in_sizes, int n_in,
                              void* d_out, int out_size, void* d_ws, size_t ws_size,
                              hipStream_t stream) {
  (void)in_sizes; (void)n_in; (void)out_size; (void)ws_size;
  const float* anchors = (const float*)d_in[1];
  const float* regr    = (const float*)d_in[2];
  const float* cls     = (const float*)d_in[3];

  float* boxes = (float*)d_ws;                       // N_ANCH*4 floats = 2.4 MB
  float* outScores = (float*)d_out;                  // [C*K]
  int*   outLabels = (int*)d_out + N_CLS * KMAX;     // [C*K] int32 bit pattern
  float* outBoxes  = (float*)d_out + 2 * N_CLS * KMAX; // [C*K,4]

  decode_kernel<<<(N_ANCH + 255) / 256, 256, 0, stream>>>(anchors, regr, boxes);
  select_nms_kernel<<<N_CLS, BLOCK, 0, stream>>>(cls, boxes, outScores,
                                                 outLabels, outBoxes);
}